// GAT_8546984919531
// MI455X (gfx1250) — compile-verified
//
#include <hip/hip_runtime.h>
#include <math.h>

typedef __attribute__((ext_vector_type(2))) float v2f;
typedef __attribute__((ext_vector_type(8))) float v8f;

#define F_IN   745
#define KPAD   748          // F_IN rounded up to multiple of 4
#define KMAIN  744          // last full unguarded K-chunk boundary
#define HID    128
#define NCLS   8
#define ROWS_PER_BLK 32
#define NEG_SLOPE 0.2f
#define EPS_PYG   1e-16f

// ---------- helpers ----------
__device__ __forceinline__ float lrelu(float x) {
    return x > 0.f ? x : NEG_SLOPE * x;
}
// monotonic float <-> uint mapping for atomicMax-based segment max
__device__ __forceinline__ unsigned fenc(float f) {
    unsigned u = __float_as_uint(f);
    return (u & 0x80000000u) ? ~u : (u | 0x80000000u);
}
__device__ __forceinline__ float fdec(unsigned u) {
    u = (u & 0x80000000u) ? (u & 0x7FFFFFFFu) : ~u;
    return __uint_as_float(u);
}

// ---------- init: zero accumulators, seed output with b2 ----------
__global__ void init_kernel(float* acc1, unsigned* m1u, float* den1,
                            unsigned* m2u, float* den2,
                            float* out, const float* b2, int N) {
    int i = blockIdx.x * blockDim.x + threadIdx.x;
    int total = N * HID;
    if (i < total) acc1[i] = 0.f;
    if (i < N) { m1u[i] = 0u; den1[i] = 0.f; m2u[i] = 0u; den2[i] = 0.f; }
    if (i < N * NCLS) out[i] = b2[i & 7];
}

// ---------- GEMM1: h1[N,128] = x[N,745] @ W1[745,128] via WMMA f32 16x16x4 ----
// Block = 32 rows x 128 cols. 8 waves; wave w owns column tile cb = 16w and
// accumulates TWO 16x16 D tiles (row panels 0-15 / 16-31) that SHARE each B
// fragment -> 2 WMMAs per B load. A panel staged once in 95.7 KB dynamic LDS.
__global__ void __launch_bounds__(256)
gemm1_kernel(const float* __restrict__ x, const float* __restrict__ W1,
             float* __restrict__ h1, int N) {
    extern __shared__ float ldsA[];            // ROWS_PER_BLK * KPAD floats
    const int nb   = blockIdx.x * ROWS_PER_BLK;
    const int tid  = threadIdx.x;
    const int lane = tid & 31;
    const int wave = tid >> 5;
    const int cb   = wave * 16;

    // cooperative stage of 32 x 745 A-panel into LDS (zero-pad K to 748)
    for (int idx = tid; idx < ROWS_PER_BLK * KPAD; idx += 256) {
        int r = idx / KPAD, k = idx % KPAD;
        int gr = nb + r;
        float v = 0.f;
        if (k < F_IN && gr < N) v = x[(long)gr * F_IN + k];
        ldsA[idx] = v;
    }
    __syncthreads();

    const int m     = lane & 15;
    const int khalf = (lane >> 4) << 1;        // 0 or 2
    const int n     = lane & 15;
    const float* ldsR0 = ldsA + m * KPAD;              // rows 0-15 panel
    const float* ldsR1 = ldsA + (m + 16) * KPAD;       // rows 16-31 panel

    v8f c0 = {}, c1 = {};
    // main loop: fully unguarded (kb+1 <= 743 < 745)
    for (int kk = 0; kk < KMAIN; kk += 4) {
        int kb = kk + khalf;
        v2f a0, a1, b;
        a0[0] = ldsR0[kb];     a0[1] = ldsR0[kb + 1];
        a1[0] = ldsR1[kb];     a1[1] = ldsR1[kb + 1];
        b[0]  = W1[(long)kb       * HID + cb + n];
        b[1]  = W1[(long)(kb + 1) * HID + cb + n];
        c0 = __builtin_amdgcn_wmma_f32_16x16x4_f32(false, a0, false, b,
                                                   (short)0, c0, false, false);
        c1 = __builtin_amdgcn_wmma_f32_16x16x4_f32(false, a1, false, b,
                                                   (short)0, c1, false, false);
    }
    // peeled tail: kk = 744, only k = 744 is a real column (LDS pad is zero)
    {
        int kb = KMAIN + khalf;                // 744 or 746
        v2f a0, a1, b;
        a0[0] = ldsR0[kb];     a0[1] = ldsR0[kb + 1];
        a1[0] = ldsR1[kb];     a1[1] = ldsR1[kb + 1];
        b[0]  = (khalf == 0) ? W1[(long)KMAIN * HID + cb + n] : 0.f;
        b[1]  = 0.f;
        c0 = __builtin_amdgcn_wmma_f32_16x16x4_f32(false, a0, false, b,
                                                   (short)0, c0, false, false);
        c1 = __builtin_amdgcn_wmma_f32_16x16x4_f32(false, a1, false, b,
                                                   (short)0, c1, false, false);
    }
    // D layout: VGPR v -> row v (lanes 0-15) / v+8 (lanes 16-31), col = lane%16
    #pragma unroll
    for (int v = 0; v < 8; ++v) {
        int r0 = nb + v + ((lane >> 4) << 3);
        int r1 = r0 + 16;
        if (r0 < N) h1[(long)r0 * HID + cb + n] = c0[v];
        if (r1 < N) h1[(long)r1 * HID + cb + n] = c1[v];
    }
}

// ---------- per-node attention logits, F=128 (wave per node) ----------
__global__ void logits128_kernel(const float* __restrict__ h,
                                 const float* __restrict__ a_src,
                                 const float* __restrict__ a_dst,
                                 float* ss, float* sd, int N) {
    int wid  = (blockIdx.x * blockDim.x + threadIdx.x) >> 5;
    int lane = threadIdx.x & 31;
    if (wid >= N) return;
    const float* row = h + (long)wid * HID;
    float s0 = 0.f, s1 = 0.f;
    #pragma unroll
    for (int j = 0; j < 4; ++j) {
        int cidx = lane + 32 * j;
        float v = row[cidx];
        s0 += v * a_src[cidx];
        s1 += v * a_dst[cidx];
    }
    #pragma unroll
    for (int off = 16; off > 0; off >>= 1) {
        s0 += __shfl_xor(s0, off, 32);
        s1 += __shfl_xor(s1, off, 32);
    }
    if (lane == 0) { ss[wid] = s0; sd[wid] = s1; }
}

// ---------- per-node logits, F=8 (thread per node) ----------
__global__ void logits8_kernel(const float* __restrict__ h,
                               const float* __restrict__ a_src,
                               const float* __restrict__ a_dst,
                               float* ss, float* sd, int N) {
    int i = blockIdx.x * blockDim.x + threadIdx.x;
    if (i >= N) return;
    float s0 = 0.f, s1 = 0.f;
    #pragma unroll
    for (int c = 0; c < NCLS; ++c) {
        float v = h[(long)i * NCLS + c];
        s0 += v * a_src[c];
        s1 += v * a_dst[c];
    }
    ss[i] = s0; sd[i] = s1;
}

// ---------- edge pass A: logit + segment max ----------
__global__ void edgeA_kernel(const int* __restrict__ src, const int* __restrict__ dst,
                             int E, int N,
                             const float* __restrict__ ss, const float* __restrict__ sd,
                             float* __restrict__ elog, unsigned* __restrict__ mu) {
    int eid = blockIdx.x * blockDim.x + threadIdx.x;
    int Et = E + N;
    if (eid >= Et) return;
    int s, d;
    if (eid < E) { s = src[eid]; d = dst[eid]; } else { s = d = eid - E; }
    float e = lrelu(ss[s] + sd[d]);
    elog[eid] = e;
    atomicMax(&mu[d], fenc(e));
}

// ---------- edge pass B: exp + segment sum ----------
__global__ void edgeB_kernel(const int* __restrict__ src, const int* __restrict__ dst,
                             int E, int N,
                             float* __restrict__ elog,
                             const unsigned* __restrict__ mu,
                             float* __restrict__ den) {
    int eid = blockIdx.x * blockDim.x + threadIdx.x;
    int Et = E + N;
    if (eid >= Et) return;
    int d = (eid < E) ? dst[eid] : (eid - E);
    float w = __expf(elog[eid] - fdec(mu[d]));
    elog[eid] = w;
    atomicAdd(&den[d], w);
}

// ---------- edge pass C, F=128: wave per edge, coalesced scatter ----------
__global__ void edgeC128_kernel(const int* __restrict__ src, const int* __restrict__ dst,
                                int E, int N,
                                const float* __restrict__ elog,
                                const float* __restrict__ den,
                                const float* __restrict__ h,
                                float* __restrict__ acc) {
    int wid  = (blockIdx.x * blockDim.x + threadIdx.x) >> 5;
    int lane = threadIdx.x & 31;
    int Et = E + N;
    if (wid >= Et) return;
    int s, d;
    if (wid < E) { s = src[wid]; d = dst[wid]; } else { s = d = wid - E; }
    float alpha = elog[wid] / (den[d] + EPS_PYG);
    const float* hr = h + (long)s * HID;
    float* ar = acc + (long)d * HID;
    #pragma unroll
    for (int j = 0; j < 4; ++j) {
        int cidx = lane + 32 * j;
        atomicAdd(&ar[cidx], alpha * hr[cidx]);
    }
}

// ---------- edge pass C, F=8: thread per edge ----------
__global__ void edgeC8_kernel(const int* __restrict__ src, const int* __restrict__ dst,
                              int E, int N,
                              const float* __restrict__ elog,
                              const float* __restrict__ den,
                              const float* __restrict__ h,
                              float* __restrict__ out) {
    int eid = blockIdx.x * blockDim.x + threadIdx.x;
    int Et = E + N;
    if (eid >= Et) return;
    int s, d;
    if (eid < E) { s = src[eid]; d = dst[eid]; } else { s = d = eid - E; }
    float alpha = elog[eid] / (den[d] + EPS_PYG);
    #pragma unroll
    for (int c = 0; c < NCLS; ++c)
        atomicAdd(&out[(long)d * NCLS + c], alpha * h[(long)s * NCLS + c]);
}

// ---------- GEMM2: h2[N,8] = relu(acc1 + b1) @ W2[128,8], WMMA, branch-free ----
__global__ void __launch_bounds__(256)
gemm2_kernel(const float* __restrict__ acc1, const float* __restrict__ b1,
             const float* __restrict__ W2, float* __restrict__ h2, int N) {
    int t    = (blockIdx.x * blockDim.x + threadIdx.x) >> 5;   // 16-row tile id
    int lane = threadIdx.x & 31;
    int ntiles = (N + 15) / 16;
    if (t >= ntiles) return;

    const int   m     = lane & 15;
    const int   khalf = (lane >> 4) << 1;
    const int   n     = lane & 15;
    const int   nc    = n & 7;                 // clamped col (always in range)
    const float nmask = (n < NCLS) ? 1.f : 0.f;
    const int   row   = t * 16 + m;
    const bool  rok   = row < N;
    const float* arow = acc1 + (long)(rok ? row : 0) * HID;
    const float rmask = rok ? 1.f : 0.f;

    v8f c = {};
    for (int kk = 0; kk < HID; kk += 4) {
        int kb = kk + khalf;
        v2f a, b;
        a[0] = fmaxf(arow[kb]     + b1[kb],     0.f) * rmask;
        a[1] = fmaxf(arow[kb + 1] + b1[kb + 1], 0.f) * rmask;
        b[0] = W2[kb * NCLS + nc]       * nmask;
        b[1] = W2[(kb + 1) * NCLS + nc] * nmask;
        c = __builtin_amdgcn_wmma_f32_16x16x4_f32(false, a, false, b,
                                                  (short)0, c, false, false);
    }
    #pragma unroll
    for (int v = 0; v < 8; ++v) {
        int r = t * 16 + v + ((lane >> 4) << 3);
        if (r < N && n < NCLS) h2[(long)r * NCLS + n] = c[v];
    }
}

// =====================================================================
extern "C" void kernel_launch(void* const* d_in, const int* in_sizes, int n_in,
                              void* d_out, int out_size, void* d_ws, size_t ws_size,
                              hipStream_t stream) {
    const float* x      = (const float*)d_in[0];
    const int*   ei     = (const int*)  d_in[1];
    const float* W1     = (const float*)d_in[2];
    const float* a1_src = (const float*)d_in[3];
    const float* a1_dst = (const float*)d_in[4];
    const float* b1     = (const float*)d_in[5];
    const float* W2     = (const float*)d_in[6];
    const float* a2_src = (const float*)d_in[7];
    const float* a2_dst = (const float*)d_in[8];
    const float* b2     = (const float*)d_in[9];
    float* out = (float*)d_out;

    const int N  = in_sizes[0] / F_IN;
    const int E  = in_sizes[1] / 2;
    const int Et = E + N;
    const int* src = ei;
    const int* dst = ei + E;

    // workspace layout (floats)
    float*    h1    = (float*)d_ws;              // N*128
    float*    acc1  = h1    + (long)N * HID;     // N*128
    float*    s1s   = acc1  + (long)N * HID;     // N
    float*    s1d   = s1s   + N;                 // N
    unsigned* m1u   = (unsigned*)(s1d + N);      // N
    float*    den1  = (float*)(m1u + N);         // N
    float*    h2    = den1  + N;                 // N*8
    float*    s2s   = h2    + (long)N * NCLS;    // N
    float*    s2d   = s2s   + N;                 // N
    unsigned* m2u   = (unsigned*)(s2d + N);      // N
    float*    den2  = (float*)(m2u + N);         // N
    float*    elog  = den2  + N;                 // Et

    const int TPB = 256;
    // init accumulators / output
    init_kernel<<<(N * HID + TPB - 1) / TPB, TPB, 0, stream>>>(
        acc1, m1u, den1, m2u, den2, out, b2, N);

    // ---- layer 1 ----
    size_t ldsBytes = (size_t)ROWS_PER_BLK * KPAD * sizeof(float);   // 95744 B
    gemm1_kernel<<<(N + ROWS_PER_BLK - 1) / ROWS_PER_BLK, TPB, ldsBytes, stream>>>(
        x, W1, h1, N);
    logits128_kernel<<<(N * 32 + TPB - 1) / TPB, TPB, 0, stream>>>(
        h1, a1_src, a1_dst, s1s, s1d, N);
    edgeA_kernel<<<(Et + TPB - 1) / TPB, TPB, 0, stream>>>(
        src, dst, E, N, s1s, s1d, elog, m1u);
    edgeB_kernel<<<(Et + TPB - 1) / TPB, TPB, 0, stream>>>(
        src, dst, E, N, elog, m1u, den1);
    edgeC128_kernel<<<((long)Et * 32 + TPB - 1) / TPB, TPB, 0, stream>>>(
        src, dst, E, N, elog, den1, h1, acc1);

    // ---- layer 2 ----
    gemm2_kernel<<<(((N + 15) / 16) * 32 + TPB - 1) / TPB, TPB, 0, stream>>>(
        acc1, b1, W2, h2, N);
    logits8_kernel<<<(N + TPB - 1) / TPB, TPB, 0, stream>>>(
        h2, a2_src, a2_dst, s2s, s2d, N);
    edgeA_kernel<<<(Et + TPB - 1) / TPB, TPB, 0, stream>>>(
        src, dst, E, N, s2s, s2d, elog, m2u);
    edgeB_kernel<<<(Et + TPB - 1) / TPB, TPB, 0, stream>>>(
        src, dst, E, N, elog, m2u, den2);
    edgeC8_kernel<<<(Et + TPB - 1) / TPB, TPB, 0, stream>>>(
        src, dst, E, N, elog, den2, h2, out);
}